// ModelNew_1580547972497
// MI455X (gfx1250) — compile-verified
//
#include <hip/hip_runtime.h>
#include <hip/hip_bf16.h>

// Problem sizes (fixed by the reference)
#define IN_F   2048
#define OUT_F  4096
#define BATCH  131072
#define RBLK   16            // row-blocks for the weight reduction (OUT_F/256)

typedef __attribute__((ext_vector_type(2))) float v2f;
typedef __attribute__((ext_vector_type(8))) float v8f;

// ---------------------------------------------------------------------------
// Stage 1: partial column-sums of weight (OUT_F x IN_F).
// grid = (IN_F/256, RBLK), block = 256. Each block sums 256 rows for its 256
// columns (fully coalesced) and writes one partial row into `partials`
// (which aliases d_out scratch — overwritten later by the GEMV kernel).
// Deterministic: fixed per-thread summation order, no atomics.
// ---------------------------------------------------------------------------
__global__ __launch_bounds__(256) void reduce_weight_p1(const float* __restrict__ w,
                                                        float* __restrict__ partials) {
    const int col = blockIdx.x * 256 + threadIdx.x;   // 0..IN_F-1
    const int r0  = blockIdx.y * (OUT_F / RBLK);
    float s = 0.0f;
    #pragma unroll 8
    for (int r = 0; r < OUT_F / RBLK; ++r) {
        s += w[(size_t)(r0 + r) * IN_F + col];
    }
    partials[(size_t)blockIdx.y * IN_F + col] = s;
}

// ---------------------------------------------------------------------------
// Stage 2: fold the RBLK partials per column in a fixed order -> ws[col].
// Also deterministic bias reduction into ws[IN_F] from block 0 via LDS tree.
// ---------------------------------------------------------------------------
__global__ __launch_bounds__(256) void reduce_weight_p2(const float* __restrict__ partials,
                                                        const float* __restrict__ bias,
                                                        float* __restrict__ ws) {
    const int col = blockIdx.x * 256 + threadIdx.x;
    float s = 0.0f;
    #pragma unroll
    for (int p = 0; p < RBLK; ++p) {
        s += partials[(size_t)p * IN_F + col];
    }
    ws[col] = s;

    // Bias sum (block 0 only): strided partials + LDS tree, fixed order.
    if (blockIdx.x == 0) {
        __shared__ float red[256];
        float bs = 0.0f;
        #pragma unroll
        for (int i = threadIdx.x; i < OUT_F; i += 256) bs += bias[i];
        red[threadIdx.x] = bs;
        __syncthreads();
        #pragma unroll
        for (int off = 128; off > 0; off >>= 1) {
            if (threadIdx.x < off) red[threadIdx.x] += red[threadIdx.x + off];
            __syncthreads();
        }
        if (threadIdx.x == 0) ws[IN_F] = red[0];
    }
}

// ---------------------------------------------------------------------------
// Stage 3: out[b] = dot(x[b,:], w_sum) + b_sum via V_WMMA_F32_16X16X4_F32.
//
// Per wave (wave32): 16 rows of x, K accumulated in chunks of 4 over 512
// chained WMMAs.
//   A (16x4 fp32, 2 VGPRs/lane): lane L<16  -> row L,   K = 4s+0, 4s+1
//                                lane L>=16 -> row L-16, K = 4s+2, 4s+3
//   => lane L reads a contiguous float2 at byte offset 16*s + 8*(L>>4) of its
//      row: a purely sequential per-lane stream; each 128B line feeds 8
//      consecutive iterations (matches the unroll-8 body).
//   B (4x16 fp32): only column N=0 nonzero = w_sum[4s..4s+3]; column lanes
//      (L&15)==0 fetch a float2 from LDS-resident w_sum, all others are 0.
//   D (16x16 fp32, 8 VGPRs): column N=0 lives in lane 0 (M=0..7) and
//      lane 16 (M=8..15); those two lanes emit the 16 results + b_sum.
// ---------------------------------------------------------------------------
__global__ __launch_bounds__(256) void gemv_wmma(const float* __restrict__ x,
                                                 const float* __restrict__ ws,
                                                 float* __restrict__ out) {
    __shared__ float s_w[IN_F];                       // 8 KB of 320 KB WGP LDS

    const int tid = threadIdx.x;
    // Coalesced one-time copy of w_sum into LDS.
    #pragma unroll
    for (int i = tid; i < IN_F; i += 256) s_w[i] = ws[i];
    __syncthreads();

    const float b_sum = ws[IN_F];

    const int lane = tid & 31;     // wave32
    const int wave = tid >> 5;     // 8 waves / block
    const int m    = lane & 15;    // row within the 16-row tile
    const int h    = lane >> 4;    // half-select: K pair 0/1 vs 2/3

    const int rowBase = (blockIdx.x * 8 + wave) * 16;

    // Per-lane A stream: row (rowBase+m), starting at K = 2*h.
    const v2f* __restrict__ px =
        (const v2f*)(x + (size_t)(rowBase + m) * IN_F) + h;
    const v2f* __restrict__ wv = (const v2f*)s_w;

    v8f c = {0.f, 0.f, 0.f, 0.f, 0.f, 0.f, 0.f, 0.f};

    #pragma unroll 8
    for (int s = 0; s < IN_F / 4; ++s) {
        v2f a = px[2 * s];                 // x[row][4s+2h .. 4s+2h+1]
        v2f bv = {0.f, 0.f};
        if (m == 0) bv = wv[2 * s + h];    // w_sum[4s+2h .. 4s+2h+1] in col 0
        // D = A*B + C   (8 args: neg_a, A, neg_b, B, c_mod, C, reuse_a, reuse_b)
        c = __builtin_amdgcn_wmma_f32_16x16x4_f32(
                false, a, false, bv, (short)0, c, false, false);
    }

    // Column N=0 of D: lane 0 -> M=0..7, lane 16 -> M=8..15.
    if (m == 0) {
        float* __restrict__ po = out + rowBase + h * 8;
        #pragma unroll
        for (int r = 0; r < 8; ++r) po[r] = c[r] + b_sum;
    }
}

// ---------------------------------------------------------------------------
// Host-side launch. Inputs: d_in[0]=x (BATCH x IN_F f32), d_in[1]=weight
// (OUT_F x IN_F f32), d_in[2]=bias (OUT_F f32). Output: BATCH f32.
// Workspace layout: ws[0..IN_F-1] = w_sum, ws[IN_F] = b_sum.
// d_out doubles as stage-1 scratch (RBLK*IN_F = 32768 floats << BATCH);
// the GEMV kernel overwrites all of d_out afterwards, so this is stateless.
// ---------------------------------------------------------------------------
extern "C" void kernel_launch(void* const* d_in, const int* in_sizes, int n_in,
                              void* d_out, int out_size, void* d_ws, size_t ws_size,
                              hipStream_t stream) {
    const float* x  = (const float*)d_in[0];
    const float* w  = (const float*)d_in[1];
    const float* b  = (const float*)d_in[2];
    float* out = (float*)d_out;
    float* ws  = (float*)d_ws;

    // Stage 1: weight partials into d_out scratch (deterministic, no atomics).
    reduce_weight_p1<<<dim3(IN_F / 256, RBLK), 256, 0, stream>>>(w, out);
    // Stage 2: fold partials + bias into ws.
    reduce_weight_p2<<<IN_F / 256, 256, 0, stream>>>(out, b, ws);

    // Stage 3: 131072 rows / (8 waves * 16 rows) = 1024 blocks.
    gemv_wmma<<<BATCH / 128, 256, 0, stream>>>(x, ws, out);
}